// BoundaryDecoderAttention_56513179681340
// MI455X (gfx1250) — compile-verified
//
#include <hip/hip_runtime.h>
#include <hip/hip_bf16.h>
#include <math.h>

// Problem constants from the reference
#define BB 64
#define TT 8192
#define DD 128
#define VPAD 130   // LDS row pitch for V_w: 130 floats -> banks (2c) distinct per 16-lane half

typedef float v2f __attribute__((ext_vector_type(2)));
typedef float v8f __attribute__((ext_vector_type(8)));

// ---------------------------------------------------------------------------
// Fk_prime[b][o] = sum_d h_tm1[b][d] * Wa_w[o][d] + Wa_b[o]
// ---------------------------------------------------------------------------
__global__ __launch_bounds__(128) void k_fkprime(const float* __restrict__ h,
                                                 const float* __restrict__ Waw,
                                                 const float* __restrict__ Wab,
                                                 float* __restrict__ Fkp) {
    int b = blockIdx.x, o = threadIdx.x;
    __shared__ float hs[DD];
    hs[o] = h[b * DD + o];
    __syncthreads();
    const float* wr = Waw + (size_t)o * DD;
    float s = Wab[o];
#pragma unroll 8
    for (int d = 0; d < DD; ++d) s = fmaf(hs[d], wr[d], s);
    Fkp[b * DD + o] = s;
}

// ---------------------------------------------------------------------------
// Scores via WMMA f32 16x16x4:
//   x[b][t] = clip( v . tanh(H[b][t] @ Vw^T + Vb + Fkp[b]) + c, -15, 15) * m
// Grid: (T/128, B). Block: 256 thr = 8 waves, each wave owns 16 rows of t.
// ---------------------------------------------------------------------------
__global__ __launch_bounds__(256) void k_scores(const float* __restrict__ H,
                                                const int*   __restrict__ mask,
                                                const float* __restrict__ Vw,
                                                const float* __restrict__ Vb,
                                                const float* __restrict__ Fkp,
                                                const float* __restrict__ vvec,
                                                const float* __restrict__ cptr,
                                                float* __restrict__ xout) {
    __shared__ float lv[DD * VPAD];           // Vw row-major, padded pitch
    const int b   = blockIdx.y;
    const int tid = threadIdx.x;

    // Stage V_w (64 KB) into LDS, coalesced.
    for (int i = tid; i < DD * DD; i += 256) {
        int o = i >> 7, d = i & 127;
        lv[o * VPAD + d] = Vw[i];
    }
    __syncthreads();

    const int wave = tid >> 5;
    const int lane = tid & 31;
    const int hi   = lane >> 4;               // K half select (ISA f32 A/B layout)
    const int c    = lane & 15;               // M for A-frag, N for B/C/D frags
    const int t0   = blockIdx.x * 128 + wave * 16;

    // A-fragment source row: M = c
    const float* hrow = H + ((size_t)b * TT + (t0 + c)) * DD;

    // Per-lane column data for the epilogue: column n = n0*16 + c
    float vfrag[8], biasf[8];
#pragma unroll
    for (int n0 = 0; n0 < 8; ++n0) {
        int n = n0 * 16 + c;
        vfrag[n0] = vvec[n];
        biasf[n0] = Vb[n] + Fkp[b * DD + n];
    }

    v8f acc[8] = {};                          // 8 N-tiles of the 16x128 row block

#pragma unroll 2
    for (int k0 = 0; k0 < 32; ++k0) {
        const int kbase = 4 * k0 + 2 * hi;    // K split across lane halves
        v2f a = *(const v2f*)(hrow + kbase);  // A: 16x4 f32 fragment
#pragma unroll
        for (int n0 = 0; n0 < 8; ++n0) {
            int n = n0 * 16 + c;
            // B[k][n] = Vw[n][k]  (Fk = H @ Vw^T)
            v2f bf = *(const v2f*)(&lv[n * VPAD + kbase]);
            acc[n0] = __builtin_amdgcn_wmma_f32_16x16x4_f32(
                false, a, false, bf, (short)0, acc[n0], false, false);
        }
    }

    // Epilogue: tanh + dot with v, per lane over its 8 rows x 8 column-tiles.
    float s[8];
#pragma unroll
    for (int j = 0; j < 8; ++j) s[j] = 0.0f;
#pragma unroll
    for (int n0 = 0; n0 < 8; ++n0) {
        float bv = biasf[n0], vv = vfrag[n0];
#pragma unroll
        for (int j = 0; j < 8; ++j)
            s[j] += vv * tanhf(acc[n0][j] + bv);
    }
    // Reduce the 16 lanes of each half (columns of each row).
#pragma unroll
    for (int j = 0; j < 8; ++j) {
        float x = s[j];
        x += __shfl_xor(x, 1, 32);
        x += __shfl_xor(x, 2, 32);
        x += __shfl_xor(x, 4, 32);
        x += __shfl_xor(x, 8, 32);
        s[j] = x;
    }
    if (c == 0) {                             // lanes 0 (rows 0-7) and 16 (rows 8-15)
        float cs = cptr[0];
#pragma unroll
        for (int j = 0; j < 8; ++j) {
            int t = t0 + j + 8 * hi;
            float sc = s[j] + cs;
            sc = fminf(fmaxf(sc, -15.0f), 15.0f);
            sc *= (float)mask[(size_t)b * TT + t];
            xout[(size_t)b * TT + t] = sc;
        }
    }
}

// ---------------------------------------------------------------------------
// In-place masked softmax over T per batch. x -> beta.
// ---------------------------------------------------------------------------
__global__ __launch_bounds__(256) void k_softmax(float* __restrict__ xb,
                                                 const int* __restrict__ mask) {
    const int b = blockIdx.x, tid = threadIdx.x;
    __shared__ float red[256];
    float* xrow = xb + (size_t)b * TT;
    const int* mrow = mask + (size_t)b * TT;

    float mx = -1e30f;
    for (int t = tid; t < TT; t += 256) mx = fmaxf(mx, xrow[t]);
    red[tid] = mx; __syncthreads();
    for (int s = 128; s > 0; s >>= 1) {
        if (tid < s) red[tid] = fmaxf(red[tid], red[tid + s]);
        __syncthreads();
    }
    mx = red[0]; __syncthreads();

    float sum = 0.0f;
    for (int t = tid; t < TT; t += 256) {
        float e = __expf(xrow[t] - mx) * (float)mrow[t];
        xrow[t] = e;
        sum += e;
    }
    red[tid] = sum; __syncthreads();
    for (int s = 128; s > 0; s >>= 1) {
        if (tid < s) red[tid] += red[tid + s];
        __syncthreads();
    }
    float inv = 1.0f / (red[0] + 1e-6f);
    __syncthreads();
    for (int t = tid; t < TT; t += 256) xrow[t] *= inv;
}

// ---------------------------------------------------------------------------
// z partials: part[b][ch][d] = sum_{t in chunk} beta[b][t] * H[b][t][d]
// Grid: (T/1024, B). 8 waves/block, wave-interleaved t; fully deterministic.
// ---------------------------------------------------------------------------
__global__ __launch_bounds__(256) void k_weighted(const float* __restrict__ H,
                                                  const float* __restrict__ beta,
                                                  float* __restrict__ part) {
    const int b = blockIdx.y, ch = blockIdx.x;
    const int wave = threadIdx.x >> 5, lane = threadIdx.x & 31;
    __shared__ float lz[8][DD];

    float a0 = 0.f, a1 = 0.f, a2 = 0.f, a3 = 0.f;
    const int tbeg = ch * 1024;
    for (int t = tbeg + wave; t < tbeg + 1024; t += 8) {
        float be = beta[(size_t)b * TT + t];           // wave-uniform broadcast
        const float* hr = H + ((size_t)b * TT + t) * DD;
        a0 = fmaf(be, hr[lane],      a0);
        a1 = fmaf(be, hr[lane + 32], a1);
        a2 = fmaf(be, hr[lane + 64], a2);
        a3 = fmaf(be, hr[lane + 96], a3);
    }
    lz[wave][lane]      = a0;
    lz[wave][lane + 32] = a1;
    lz[wave][lane + 64] = a2;
    lz[wave][lane + 96] = a3;
    __syncthreads();
    if (threadIdx.x < DD) {
        float s = 0.f;
#pragma unroll
        for (int w = 0; w < 8; ++w) s += lz[w][threadIdx.x];
        part[((size_t)b * 8 + ch) * DD + threadIdx.x] = s;
    }
}

__global__ __launch_bounds__(128) void k_reduce_z(const float* __restrict__ part,
                                                  float* __restrict__ z) {
    const int b = blockIdx.x, d = threadIdx.x;
    float s = 0.f;
#pragma unroll
    for (int ch = 0; ch < 8; ++ch) s += part[((size_t)b * 8 + ch) * DD + d];
    z[b * DD + d] = s;
}

// ---------------------------------------------------------------------------
extern "C" void kernel_launch(void* const* d_in, const int* in_sizes, int n_in,
                              void* d_out, int out_size, void* d_ws, size_t ws_size,
                              hipStream_t stream) {
    const float* H    = (const float*)d_in[0];   // [B,T,D]
    const int*   mask = (const int*)  d_in[1];   // [B,T]
    const float* h1   = (const float*)d_in[2];   // [B,D]
    const float* Vw   = (const float*)d_in[3];   // [D,D]
    const float* Vb   = (const float*)d_in[4];   // [D]
    const float* Waw  = (const float*)d_in[5];   // [D,D]
    const float* Wab  = (const float*)d_in[6];   // [D]
    const float* vv   = (const float*)d_in[7];   // [D]
    const float* c    = (const float*)d_in[8];   // [1]

    float* z    = (float*)d_out;                 // [B,D]
    float* beta = z + BB * DD;                   // [B,T] (also holds x, then e)

    float* Fkp  = (float*)d_ws;                  // [B,D]      32 KB
    float* part = Fkp + BB * DD;                 // [B,8,D]   256 KB

    k_fkprime<<<BB, 128, 0, stream>>>(h1, Waw, Wab, Fkp);

    dim3 gs(TT / 128, BB);
    k_scores<<<gs, 256, 0, stream>>>(H, mask, Vw, Vb, Fkp, vv, c, beta);

    k_softmax<<<BB, 256, 0, stream>>>(beta, mask);

    dim3 gw(TT / 1024, BB);
    k_weighted<<<gw, 256, 0, stream>>>(H, beta, part);

    k_reduce_z<<<BB, 128, 0, stream>>>(part, z);
}